// MultiScaleFusionRoIAlign_72258529788127
// MI455X (gfx1250) — compile-verified
//
#include <hip/hip_runtime.h>

typedef __attribute__((ext_vector_type(2))) float v2f;
typedef __attribute__((ext_vector_type(8))) float v8f;

__device__ __forceinline__ int clampi(int v, int lo, int hi) {
    return v < lo ? lo : (v > hi ? hi : v);
}

constexpr int BATCH = 2, NROI = 256, CPS = 256;   // channels per scale
constexpr int KSZ[3]  = {24, 16, 8};              // patch side per scale (K dim, padded)
constexpr int HWS[3]  = {100, 50, 25};            // source H=W per scale
constexpr int AOFF[3] = {0, 384, 640};            // offsets of 16-row-padded A mats in sA

__global__ __launch_bounds__(256)
void roialign_sep_wmma(const float* __restrict__ x0,
                       const float* __restrict__ x1,
                       const float* __restrict__ x2,
                       const float* __restrict__ boxes,
                       const float* __restrict__ ratio,
                       const float* __restrict__ offtl,
                       float* __restrict__ out)
{
    __shared__ float sP[24 * 24 * 16];   // patch [y][x][c16]
    __shared__ float sQ[24 * 7 * 16];    // Q [x][ph][c16]
    __shared__ float sOb[16 * 49];       // out chunk [c16][ph][pw]
    __shared__ float sA[2][768];         // axis(0=y,1=x): 3 scales of 16xK weight mats
    __shared__ int   sLo[2][14], sHi[2][14], sVal[2][14];
    __shared__ float sFr[2][14];
    __shared__ int   sBase0[2];
    __shared__ int   sBaseS[2][3];

    const int tid  = threadIdx.x;
    const int lane = tid & 31, wave = tid >> 5;
    const int half = lane >> 4, l16 = lane & 15;
    const int roi  = blockIdx.x;          // = b*NROI + n
    const int bb   = roi >> 8;            // batch index (NROI==256)

    // zero weight matrices
    for (int t = tid; t < 2 * 768; t += 256) (&sA[0][0])[t] = 0.f;

    // ---- per-axis sample metadata (axis 0 = y uses rH/top, axis 1 = x uses rW/left) ----
    if (tid < 28) {
        int a = tid / 14, i = tid % 14;
        float rr = ratio[bb * 2 + (a ? 1 : 0)];
        float of = offtl[bb * 2 + (a ? 1 : 0)];
        float c1 = boxes[roi * 4 + (a ? 0 : 1)] * rr + of;
        float c2 = boxes[roi * 4 + (a ? 2 : 3)] * rr + of;
        float scale = 100.f / (1250.f * ratio[0] + 2.f * offtl[0]);
        float s1 = c1 * scale, s2 = c2 * scale;
        float bsz = fmaxf(s2 - s1, 1.f) * (1.f / 7.f);
        float g  = (float)(i >> 1) + ((float)(i & 1) + 0.5f) * 0.5f;
        float cs = s1 + g * bsz;
        sVal[a][i] = (cs >= -1.f) && (cs <= 100.f);
        float cc = fminf(fmaxf(cs, 0.f), 99.f);
        int lo = (int)floorf(cc);
        sLo[a][i] = lo;
        sHi[a][i] = min(lo + 1, 99);
        sFr[a][i] = cc - (float)lo;
        if (i == 0) sBase0[a] = lo;
    }
    __syncthreads();
    if (tid < 2) {
        int b0 = sBase0[tid];
        sBaseS[tid][0] = b0;
        sBaseS[tid][1] = clampi((int)floorf((float)b0 * 0.5f  - 0.25f),  0, 49);
        sBaseS[tid][2] = clampi((int)floorf((float)b0 * 0.25f - 0.375f), 0, 24);
    }
    __syncthreads();
    // ---- accumulate folded weights: (0.5 * roi-bilinear) composed with 2x/4x upsample ----
    if (tid < 28) {
        int a = tid / 14, i = tid % 14;
        if (sVal[a][i]) {
            int ph = i >> 1;
            float fr = sFr[a][i];
            for (int t = 0; t < 2; t++) {
                int   r = t ? sHi[a][i] : sLo[a][i];      // fused-grid row/col in [0,99]
                float w = 0.5f * (t ? fr : 1.f - fr);
                // scale 0: identity mapping
                int idx = r - sBaseS[a][0];
                if (idx >= 0 && idx < 24) atomicAdd(&sA[a][ph * 24 + idx], w);
                // scale 1: src = r/2 - 0.25 (half-pixel 2x), edge-clamped (== jax resize)
                {
                    float sy = (float)r * 0.5f - 0.25f;
                    float fl = floorf(sy), f2 = sy - fl;
                    int l = clampi((int)fl, 0, 49), h = clampi((int)fl + 1, 0, 49);
                    int i1 = l - sBaseS[a][1], i2 = h - sBaseS[a][1];
                    if (i1 >= 0 && i1 < 16) atomicAdd(&sA[a][384 + ph * 16 + i1], w * (1.f - f2));
                    if (i2 >= 0 && i2 < 16) atomicAdd(&sA[a][384 + ph * 16 + i2], w * f2);
                }
                // scale 2: src = r/4 - 0.375 (half-pixel 4x)
                {
                    float sy = (float)r * 0.25f - 0.375f;
                    float fl = floorf(sy), f2 = sy - fl;
                    int l = clampi((int)fl, 0, 24), h = clampi((int)fl + 1, 0, 24);
                    int i1 = l - sBaseS[a][2], i2 = h - sBaseS[a][2];
                    if (i1 >= 0 && i1 < 8) atomicAdd(&sA[a][640 + ph * 8 + i1], w * (1.f - f2));
                    if (i2 >= 0 && i2 < 8) atomicAdd(&sA[a][640 + ph * 8 + i2], w * f2);
                }
            }
        }
    }
    __syncthreads();

    #pragma unroll
    for (int s = 0; s < 3; s++) {
        const float* __restrict__ feat = (s == 0) ? x0 : ((s == 1) ? x1 : x2);
        const int K = KSZ[s], HW = HWS[s], aoff = AOFF[s], nk = K >> 2;
        const int by = sBaseS[0][s], bx = sBaseS[1][s];
        for (int chunk = 0; chunk < 16; chunk++) {
            const int cbase = chunk * 16;
            // prefetch next channel chunk (global_prefetch_b8)
            if (cbase + 16 < CPS) {
                const float* pf =
                    feat + (((size_t)bb * CPS + cbase + 16 + (tid & 15)) * HW + by) * HW + bx;
                __builtin_prefetch(pf, 0, 1);
            }
            // ---- stage patch into LDS, coalesced along x; layout sP[y][x][c] ----
            const int total = K * K * 16;
            for (int t = tid; t < total; t += 256) {
                int x = t % K; int rem = t / K; int c = rem & 15; int y = rem >> 4;
                int gy = min(by + y, HW - 1), gx = min(bx + x, HW - 1);
                sP[(y * K + x) * 16 + c] =
                    feat[(((size_t)bb * CPS + cbase + c) * HW + gy) * HW + gx];
            }
            __syncthreads();
            // ---- step 1: Q[x][ph][c] = sum_y Ay[ph][y] * P[y][x][c] ----
            {
                v2f af[6];
                #pragma unroll
                for (int kk = 0; kk < 6; kk++) {
                    if (kk < nk) {
                        int k0 = 4 * kk + 2 * half;           // A 16x4 f32 layout: v0=K0/K2, v1=K1/K3
                        af[kk].x = sA[0][aoff + l16 * K + k0];
                        af[kk].y = sA[0][aoff + l16 * K + k0 + 1];
                    }
                }
                for (int xt = wave; xt < K; xt += 8) {        // K in {24,16,8}: uniform per wave
                    v8f acc = {};
                    for (int kk = 0; kk < nk; kk++) {
                        int k0 = 4 * kk + 2 * half;
                        v2f bf;
                        bf.x = sP[(k0 * K + xt) * 16 + l16];  // B[k][n=c]: v0=K0/K2, v1=K1/K3
                        bf.y = sP[((k0 + 1) * K + xt) * 16 + l16];
                        acc = __builtin_amdgcn_wmma_f32_16x16x4_f32(
                                  false, af[kk], false, bf, (short)0, acc, false, false);
                    }
                    if (half == 0) {                           // rows M=0..6 live in lanes 0-15
                        #pragma unroll
                        for (int v = 0; v < 7; v++) sQ[(xt * 7 + v) * 16 + l16] = acc[v];
                    }
                }
            }
            __syncthreads();
            // ---- step 2: out[pw][ph][c] = sum_x Ax[pw][x] * Q[x][ph][c] ----
            if (wave < 7) {
                v2f af[6];
                #pragma unroll
                for (int kk = 0; kk < 6; kk++) {
                    if (kk < nk) {
                        int k0 = 4 * kk + 2 * half;
                        af[kk].x = sA[1][aoff + l16 * K + k0];
                        af[kk].y = sA[1][aoff + l16 * K + k0 + 1];
                    }
                }
                v8f acc = {};
                for (int kk = 0; kk < nk; kk++) {
                    int k0 = 4 * kk + 2 * half;
                    v2f bf;
                    bf.x = sQ[(k0 * 7 + wave) * 16 + l16];
                    bf.y = sQ[((k0 + 1) * 7 + wave) * 16 + l16];
                    acc = __builtin_amdgcn_wmma_f32_16x16x4_f32(
                              false, af[kk], false, bf, (short)0, acc, false, false);
                }
                if (half == 0) {
                    #pragma unroll
                    for (int v = 0; v < 7; v++) sOb[l16 * 49 + wave * 7 + v] = acc[v];
                }
            }
            __syncthreads();
            // ---- coalesced writeback: 16 channels x 49 are contiguous in the output ----
            {
                size_t obase = (((size_t)roi) * 768 + (size_t)(s * 256 + cbase)) * 49;
                for (int t = tid; t < 784; t += 256) out[obase + t] = sOb[t];
            }
            __syncthreads();
        }
    }
}

extern "C" void kernel_launch(void* const* d_in, const int* in_sizes, int n_in,
                              void* d_out, int out_size, void* d_ws, size_t ws_size,
                              hipStream_t stream)
{
    const float* x0    = (const float*)d_in[0];
    const float* x1    = (const float*)d_in[1];
    const float* x2    = (const float*)d_in[2];
    const float* boxes = (const float*)d_in[3];
    const float* ratio = (const float*)d_in[4];
    const float* offtl = (const float*)d_in[5];
    float* outp = (float*)d_out;

    roialign_sep_wmma<<<dim3(BATCH * NROI), dim3(256), 0, stream>>>(
        x0, x1, x2, boxes, ratio, offtl, outp);
}